// FastFoodTransform_80564996539217
// MI455X (gfx1250) — compile-verified
//
#include <hip/hip_runtime.h>

typedef __attribute__((ext_vector_type(2))) float v2f;
typedef __attribute__((ext_vector_type(4))) float v4f;
typedef __attribute__((ext_vector_type(8))) float v8f;

#define FF_DIM   4096
#define FF_TPB   512                  // 16 waves (wave32): one 16-col tile per wave per stage
#define FF_SCALE 0.015625f            // 1/sqrt(4096)

// Pad every 16 floats by 1 (stride 17, coprime with 64 banks) -> all three
// FWHT stage access patterns are LDS-bank-conflict free.
__device__ __forceinline__ int ffpad(int e) { return e + (e >> 4); }

__global__ __launch_bounds__(FF_TPB) void fastfood_kernel(
    const float* __restrict__ x, const float* __restrict__ d1,
    const float* __restrict__ d2, const float* __restrict__ gauss_d,
    const int* __restrict__ perm, float* __restrict__ out)
{
    __shared__ float buf[FF_DIM + (FF_DIM >> 4)];   // 4352 floats = 17 KB

    const int tid  = threadIdx.x;
    const int lane = tid & 31;
    const int wv   = tid >> 5;        // wave id 0..15
    const int half = lane >> 4;       // lane half selects K/M sub-block
    const int t    = lane & 15;       // column / M-row within fragment

    const float* xr = x   + (size_t)blockIdx.x * FF_DIM;
    float*       yr = out + (size_t)blockIdx.x * FF_DIM;

    // ---- phase 0: stream row in (NT), apply d1, stage into padded LDS ----
#pragma unroll
    for (int u = 0; u < 2; ++u) {
        int i4 = tid + u * FF_TPB;                       // 1024 float4 per row
        v4f v = __builtin_nontemporal_load((const v4f*)xr + i4);
        v4f s = ((const v4f*)d1)[i4];
        int p = ffpad(i4 * 4);                           // e%16==0..12, same 16-group
        buf[p + 0] = v.x * s.x;
        buf[p + 1] = v.y * s.y;
        buf[p + 2] = v.z * s.z;
        buf[p + 3] = v.w * s.w;
    }

    // ---- A fragments of constant H16 (shared by all 6 stages) ----
    // A 16x4 fp32 layout: lane<16 -> K=0,1 ; lane>=16 -> K=2,3 ; M = lane%16
    v2f hA[4];
#pragma unroll
    for (int c = 0; c < 4; ++c) {
        int k0 = 4 * c + 2 * half;
        hA[c].x = (__builtin_popcount(t & (k0    )) & 1) ? -1.0f : 1.0f;
        hA[c].y = (__builtin_popcount(t & (k0 + 1)) & 1) ? -1.0f : 1.0f;
    }

    __syncthreads();

    // One H16 left-multiplication along axis of stride aS; this wave's 16-col
    // tile starts at `base`, columns advance by tS. In-place: the tile's read
    // and write sets are identical and wave-private.
    auto stage = [&](int base, int aS, int tS) {
        const int tOff = base + t * tS;
        v8f acc = {0.f, 0.f, 0.f, 0.f, 0.f, 0.f, 0.f, 0.f};
#pragma unroll
        for (int c = 0; c < 4; ++c) {
            int k0 = 4 * c + 2 * half;                   // B 4x16: rows 0,1 | 2,3 per half
            v2f b;
            b.x = buf[ffpad(tOff + (k0    ) * aS)];
            b.y = buf[ffpad(tOff + (k0 + 1) * aS)];
            acc = __builtin_amdgcn_wmma_f32_16x16x4_f32(
                false, hA[c], false, b, (short)0, acc, false, false);
        }
#pragma unroll
        for (int r = 0; r < 8; ++r)                      // C/D: rows r | r+8 per half
            buf[ffpad(tOff + (r + 8 * half) * aS)] = acc[r];
        __syncthreads();
    };

    // ---- FWHT #1 :  H16 along strides 1, 16, 256 ----
    stage(wv * 256, 1, 16);
    stage(wv * 256, 16, 1);
    stage(wv * 16, 256, 1);

    // ---- permute + gauss diagonal (register-buffered, in-place) ----
    float z[8];
#pragma unroll
    for (int u = 0; u < 8; ++u) {
        int i = tid + u * FF_TPB;
        z[u] = buf[ffpad(perm[i])] * gauss_d[i];
    }
    __syncthreads();
#pragma unroll
    for (int u = 0; u < 8; ++u)
        buf[ffpad(tid + u * FF_TPB)] = z[u];
    __syncthreads();

    // ---- FWHT #2 ----
    stage(wv * 256, 1, 16);
    stage(wv * 256, 16, 1);
    stage(wv * 16, 256, 1);

    // ---- final: d2 * SCALE, stream out (NT) ----
#pragma unroll
    for (int u = 0; u < 2; ++u) {
        int i4 = tid + u * FF_TPB;
        int p = ffpad(i4 * 4);
        v4f s = ((const v4f*)d2)[i4];
        v4f v;
        v.x = buf[p + 0] * s.x * FF_SCALE;
        v.y = buf[p + 1] * s.y * FF_SCALE;
        v.z = buf[p + 2] * s.z * FF_SCALE;
        v.w = buf[p + 3] * s.w * FF_SCALE;
        __builtin_nontemporal_store(v, (v4f*)yr + i4);
    }
}

extern "C" void kernel_launch(void* const* d_in, const int* in_sizes, int n_in,
                              void* d_out, int out_size, void* d_ws, size_t ws_size,
                              hipStream_t stream) {
    (void)in_sizes; (void)n_in; (void)out_size; (void)d_ws; (void)ws_size;
    const float* x       = (const float*)d_in[0];
    const float* d1      = (const float*)d_in[1];
    const float* d2      = (const float*)d_in[2];
    const float* gauss_d = (const float*)d_in[3];
    const int*   perm    = (const int*)d_in[4];   // jnp int64 w/o x64 -> int32
    float*       out     = (float*)d_out;

    fastfood_kernel<<<8192, FF_TPB, 0, stream>>>(x, d1, d2, gauss_d, perm, out);
}